// LNLSelfAttention_33509334843620
// MI455X (gfx1250) — compile-verified
//
#include <hip/hip_runtime.h>
#include <hip/hip_bf16.h>

typedef __attribute__((ext_vector_type(2))) float v2f;
typedef __attribute__((ext_vector_type(8))) float v8f;
typedef __attribute__((ext_vector_type(4))) int   v4i;

#define BN_TOK 1024
#define E_DIM  768
#define H_NUM  12
#define HD_DIM 64

#define KT    32            // K-tile

#define AS1 __attribute__((address_space(1)))
#define AS3 __attribute__((address_space(3)))

#if defined(__gfx1250__) && \
    __has_builtin(__builtin_amdgcn_global_load_async_to_lds_b128) && \
    __has_builtin(__builtin_amdgcn_s_wait_asynccnt)
#define USE_ASYNC_LDS 1
#else
#define USE_ASYNC_LDS 0
#endif

__device__ __forceinline__ void stage16(const float* __restrict__ gp, float* lp)
{
#if USE_ASYNC_LDS
    __builtin_amdgcn_global_load_async_to_lds_b128((AS1 v4i*)gp, (AS3 v4i*)lp, 0, 0);
#else
    const float4 v = *(const float4*)gp;
    lp[0] = v.x; lp[1] = v.y; lp[2] = v.z; lp[3] = v.w;
#endif
}

// ---------------------------------------------------------------------------
// Generic f32 WMMA GEMM: C[M,N] = epilogue( A[M,K] * B[K,N] )
//   WS = wave-subtile count per dim: each wave computes (16*WS)x(16*WS),
//        block = 8 waves (2x4) -> tile (32*WS) x (64*WS).
//   A row-major (lda); B row-major (ldb) unless TRANS_B (Bmem is [N,K]).
//   blockIdx.z = head; aHS/bHS/cHS per-head element strides.
//   mode 0: (acc + bias[col]) * scale            (QKV projection)
//   mode 1: g/mask-gated exp(acc)                (attention scores)
//   mode 2: acc                                  (power series / attn@v)
// ---------------------------------------------------------------------------
template<bool TRANS_B, int WS>
__global__ __launch_bounds__(256)
void wmma_gemm_f32(const float* __restrict__ A, int lda, long aHS,
                   const float* __restrict__ Bm, int ldb, long bHS,
                   float* __restrict__ C, int ldc, long cHS,
                   int K, int mode, float scale,
                   const float* __restrict__ bias,
                   const int* __restrict__ g,
                   const float* __restrict__ amask)
{
    constexpr int BM  = 32 * WS;
    constexpr int BN  = 64 * WS;
    constexpr int LPA = KT + 4;      // 36 floats: 16B-aligned rows, conflict-free
    constexpr int LPN = BN + 4;      // 68 / 132 floats

    __shared__ float sA[BM * LPA];   // [m][k]
    __shared__ float sB[KT * LPN];   // [k][n]

    const int h = blockIdx.z;
    A  += (long)h * aHS;
    Bm += (long)h * bHS;
    C  += (long)h * cHS;

    const int br = blockIdx.y * BM;
    const int bc = blockIdx.x * BN;
    const int t    = threadIdx.x;
    const int wave = t >> 5;
    const int lane = t & 31;
    const int wrow = (wave >> 2) * (16 * WS);   // 0 / 16*WS
    const int wcol = (wave & 3)  * (16 * WS);   // 0..3 * 16*WS
    const int lrow = lane & 15;
    const int koff = (lane >> 4) << 1;          // 0 or 2

    const v8f vzero = {0.f, 0.f, 0.f, 0.f, 0.f, 0.f, 0.f, 0.f};
    v8f acc[WS][WS];
    #pragma unroll
    for (int tr = 0; tr < WS; ++tr)
        #pragma unroll
        for (int tc = 0; tc < WS; ++tc) acc[tr][tc] = vzero;

    for (int k0 = 0; k0 < K; k0 += KT) {
        __syncthreads();
        // --- stage A tile (BM x KT): b128 per thread per 32-row slab -------
        {
            const int r0 = t >> 3;            // 0..31
            const int c4 = (t & 7) << 2;      // 0..28
            #pragma unroll
            for (int rr = 0; rr < BM; rr += 32) {
                stage16(A + (long)(br + r0 + rr) * lda + k0 + c4,
                        &sA[(r0 + rr) * LPA + c4]);
            }
        }
        // --- stage B tile into sB[k][n] ------------------------------------
        if (TRANS_B) {
            // Bmem is [N,K] row-major: read along k, transpose into LDS
            const int n0 = t >> 3;            // 0..31
            const int c4 = (t & 7) << 2;
            #pragma unroll
            for (int nn = 0; nn < BN; nn += 32) {
                const float4 vb = *(const float4*)(Bm + (long)(bc + n0 + nn) * ldb + k0 + c4);
                sB[(c4 + 0) * LPN + n0 + nn] = vb.x;
                sB[(c4 + 1) * LPN + n0 + nn] = vb.y;
                sB[(c4 + 2) * LPN + n0 + nn] = vb.z;
                sB[(c4 + 3) * LPN + n0 + nn] = vb.w;
            }
        } else {
            // Bmem is [K,N] row-major: contiguous 16B copies
            constexpr int NT4   = BN / 4;       // threads per k-row
            constexpr int KSTEP = 256 / NT4;
            const int kk0 = t / NT4;
            const int n4  = (t % NT4) << 2;
            #pragma unroll
            for (int kk = kk0; kk < KT; kk += KSTEP) {
                stage16(Bm + (long)(k0 + kk) * ldb + bc + n4,
                        &sB[kk * LPN + n4]);
            }
        }
#if USE_ASYNC_LDS
        __builtin_amdgcn_s_wait_asynccnt(0);
#endif
        __syncthreads();

        // --- compute: WS*WS x V_WMMA_F32_16X16X4_F32 per 4-deep K step -----
        // A frag: lanes 0-15 -> M=lane, K pair {k,k+1}; lanes 16-31 -> {k+2,k+3}
        #pragma unroll
        for (int kk = 0; kk < KT; kk += 4) {
            v2f af[WS], bf[WS];
            #pragma unroll
            for (int s = 0; s < WS; ++s) {
                const float* pa = &sA[(wrow + s * 16 + lrow) * LPA + kk + koff];
                af[s].x = pa[0];
                af[s].y = pa[1];
                const float* pb = &sB[(kk + koff) * LPN + wcol + s * 16 + lrow];
                bf[s].x = pb[0];
                bf[s].y = pb[LPN];
            }
            #pragma unroll
            for (int tr = 0; tr < WS; ++tr)
                #pragma unroll
                for (int tc = 0; tc < WS; ++tc)
                    acc[tr][tc] = __builtin_amdgcn_wmma_f32_16x16x4_f32(
                        false, af[tr], false, bf[tc], (short)0, acc[tr][tc],
                        false, false);
        }
    }

    // --- epilogue ----------------------------------------------------------
    // C/D layout: lanes 0-15: VGPR r -> M=r ; lanes 16-31: VGPR r -> M=8+r
    #pragma unroll
    for (int tr = 0; tr < WS; ++tr) {
        #pragma unroll
        for (int tc = 0; tc < WS; ++tc) {
            const int ncol  = bc + wcol + tc * 16 + lrow;
            const int mbase = br + wrow + tr * 16 + ((lane >> 4) << 3);
            #pragma unroll
            for (int r2 = 0; r2 < 8; ++r2) {
                const int gi = mbase + r2;
                float val = acc[tr][tc][r2];
                if (mode == 0) {
                    val = (val + bias[ncol]) * scale;
                } else if (mode == 1) {
                    const int  ge = g[(long)gi * BN_TOK + ncol];
                    const bool bm = (amask[gi] >= 0.f) && (amask[ncol] >= 0.f);
                    val = (bm && ge > 0) ? __expf(val) : 0.f;
                }
                C[(long)gi * ldc + ncol] = val;
            }
        }
    }
}

// ---------------------------------------------------------------------------
// out_degree[row] = sum_j W_G[row, j]      (row = h*BN + i, n = BN)
// ---------------------------------------------------------------------------
__global__ __launch_bounds__(256)
void rowsum_kernel(const float* __restrict__ W, float* __restrict__ deg, int n)
{
    __shared__ float red[256];
    const long row = blockIdx.x;
    const float* p = W + row * (long)n;
    float s = 0.f;
    for (int j = threadIdx.x; j < n; j += 256) s += p[j];
    red[threadIdx.x] = s;
    __syncthreads();
    for (int off = 128; off > 0; off >>= 1) {
        if (threadIdx.x < off) red[threadIdx.x] += red[threadIdx.x + off];
        __syncthreads();
    }
    if (threadIdx.x == 0) deg[row] = red[0];
}

// rhos[h] = max_i deg[h*n + i]
__global__ __launch_bounds__(256)
void rho_kernel(const float* __restrict__ deg, float* __restrict__ rhos, int n)
{
    __shared__ float red[256];
    const int h = blockIdx.x;
    float m = -1e30f;
    for (int i = threadIdx.x; i < n; i += 256) m = fmaxf(m, deg[(long)h * n + i]);
    red[threadIdx.x] = m;
    __syncthreads();
    for (int off = 128; off > 0; off >>= 1) {
        if (threadIdx.x < off) red[threadIdx.x] = fmaxf(red[threadIdx.x], red[threadIdx.x + off]);
        __syncthreads();
    }
    if (threadIdx.x == 0) rhos[h] = red[0];
}

// Bmat = W_G + (rho[h] - deg[i]) on diag ; B_power = I ; L_gamma = I
__global__ __launch_bounds__(256)
void bmat_init_kernel(const float* __restrict__ WG, const float* __restrict__ deg,
                      const float* __restrict__ rhos,
                      float* __restrict__ Bmat, float* __restrict__ Bp,
                      float* __restrict__ Lg, int n)
{
    const long idx = (long)blockIdx.x * 256 + threadIdx.x;
    const int j = (int)(idx % n);
    const long t = idx / n;
    const int i = (int)(t % n);
    const int h = (int)(t / n);
    const float e = (i == j) ? 1.f : 0.f;
    Bmat[idx] = WG[idx] + ((i == j) ? (rhos[h] - deg[(long)h * n + i]) : 0.f);
    Bp[idx]   = e;
    Lg[idx]   = e;
}

// L_gamma += (c * rho^-ii) * B_power
__global__ __launch_bounds__(256)
void axpy_pow_kernel(const float* __restrict__ Bp, const float* __restrict__ rhos,
                     float* __restrict__ Lg, float c, int ii, int n)
{
    const long idx = (long)blockIdx.x * 256 + threadIdx.x;
    const int h = (int)(idx / ((long)n * n));
    const float rinv = 1.f / rhos[h];
    float p = 1.f;
    for (int q = 0; q < ii; ++q) p *= rinv;
    Lg[idx] += (c * p) * Bp[idx];
}

// rsL[row] = rho^0.5 * sum_{j != i} |L_gamma[row, j]|
__global__ __launch_bounds__(256)
void lrow_kernel(const float* __restrict__ Lg, const float* __restrict__ rhos,
                 float* __restrict__ rsL, int n)
{
    __shared__ float red[256];
    const long row = blockIdx.x;
    const int  i = (int)(row % n);
    const int  h = (int)(row / n);
    const float* p = Lg + row * (long)n;
    float s = 0.f;
    for (int j = threadIdx.x; j < n; j += 256)
        if (j != i) s += fabsf(p[j]);
    red[threadIdx.x] = s;
    __syncthreads();
    for (int off = 128; off > 0; off >>= 1) {
        if (threadIdx.x < off) red[threadIdx.x] += red[threadIdx.x + off];
        __syncthreads();
    }
    if (threadIdx.x == 0) rsL[row] = sqrtf(rhos[h]) * red[0];
}

// comb = eps * W_G/deg  + (1-eps) * l1norm(diag - rho^g * Lg)
// (diagonal of the second term is exactly zero)
__global__ __launch_bounds__(256)
void combine_kernel(const float* __restrict__ WG, const float* __restrict__ Lg,
                    const float* __restrict__ deg, const float* __restrict__ rhos,
                    const float* __restrict__ rsL, float* __restrict__ comb, int n)
{
    const long idx = (long)blockIdx.x * 256 + threadIdx.x;
    const int j = (int)(idx % n);
    const long t = idx / n;
    const int i = (int)(t % n);
    const int h = (int)(t / n);
    const float wgn = WG[idx] / fmaxf(deg[(long)h * n + i], 1e-12f);
    const float sc  = sqrtf(rhos[h]);                 // rho^GAMMA, GAMMA=0.5
    const float m   = (i == j) ? 0.f : (-sc * Lg[idx]);
    const float ln  = m / fmaxf(rsL[(long)h * n + i], 1e-12f);
    comb[idx] = 0.8f * wgn + 0.2f * ln;               // LOCAL_EPS = 0.8
}

// ---------------------------------------------------------------------------
extern "C" void kernel_launch(void* const* d_in, const int* in_sizes, int n_in,
                              void* d_out, int out_size, void* d_ws, size_t ws_size,
                              hipStream_t stream)
{
    (void)in_sizes; (void)n_in; (void)out_size; (void)ws_size;
    const float* x   = (const float*)d_in[0];
    const float* Wq  = (const float*)d_in[1];
    const float* bq  = (const float*)d_in[2];
    const float* Wk  = (const float*)d_in[3];
    const float* bk  = (const float*)d_in[4];
    const float* Wv  = (const float*)d_in[5];
    const float* bv  = (const float*)d_in[6];
    const int*   g   = (const int*)d_in[7];
    const float* am  = (const float*)d_in[8];
    float* out = (float*)d_out;

    const long QKV  = (long)BN_TOK * E_DIM;          //    786,432 floats
    const long MAT  = (long)H_NUM * BN_TOK * BN_TOK; // 12,582,912 floats
    float* ws   = (float*)d_ws;
    float* q    = ws;
    float* k    = q    + QKV;
    float* v    = k    + QKV;
    float* WG   = v    + QKV;
    float* Bmat = WG   + MAT;   // later reused for the combined attention matrix
    float* Bp0  = Bmat + MAT;
    float* Bp1  = Bp0  + MAT;
    float* Lg   = Bp1  + MAT;
    float* deg  = Lg   + MAT;                    // H*BN
    float* rhos = deg  + (long)H_NUM * BN_TOK;   // H
    float* rsL  = rhos + H_NUM;                  // H*BN

    const dim3 blk(256);
    const float qscale = 0.125f;                 // 1/sqrt(HD)

    // 1) QKV projections: [1024,768] @ [768,768] + bias  (64x128 block tiles)
    {
        dim3 grd(E_DIM / 128, BN_TOK / 64, 1);
        wmma_gemm_f32<false, 2><<<grd, blk, 0, stream>>>(x, E_DIM, 0, Wq, E_DIM, 0,
            q, E_DIM, 0, E_DIM, 0, qscale, bq, nullptr, nullptr);
        wmma_gemm_f32<false, 2><<<grd, blk, 0, stream>>>(x, E_DIM, 0, Wk, E_DIM, 0,
            k, E_DIM, 0, E_DIM, 0, 1.0f, bk, nullptr, nullptr);
        wmma_gemm_f32<false, 2><<<grd, blk, 0, stream>>>(x, E_DIM, 0, Wv, E_DIM, 0,
            v, E_DIM, 0, E_DIM, 0, 1.0f, bv, nullptr, nullptr);
    }

    // 2) W_G[h] = mask( exp( K_h @ Q_h^T ) )   (K=64, B transposed access)
    {
        dim3 grd(BN_TOK / 128, BN_TOK / 64, H_NUM);
        wmma_gemm_f32<true, 2><<<grd, blk, 0, stream>>>(k, E_DIM, HD_DIM, q, E_DIM, HD_DIM,
            WG, BN_TOK, (long)BN_TOK * BN_TOK, HD_DIM, 1, 1.0f, nullptr, g, am);
    }

    // 3) degrees + rhos
    rowsum_kernel<<<dim3(H_NUM * BN_TOK), blk, 0, stream>>>(WG, deg, BN_TOK);
    rho_kernel<<<dim3(H_NUM), blk, 0, stream>>>(deg, rhos, BN_TOK);

    // 4) Bmat, B_power = I, L_gamma = I
    const long NEL = MAT;
    bmat_init_kernel<<<dim3((unsigned)(NEL / 256)), blk, 0, stream>>>(
        WG, deg, rhos, Bmat, Bp0, Lg, BN_TOK);

    // 5) 7-term power series: B_power = Bmat @ B_power ; L_gamma += coef*B_power
    float* cur = Bp0;
    float* nxt = Bp1;
    float num = 1.0f, fact = 1.0f;
    for (int ii = 1; ii <= 7; ++ii) {
        num  = num * (0.5f - (float)ii + 1.0f) * -1.0f;   // GAMMA = 0.5
        fact *= (float)ii;
        const float c = num / fact;                       // coef = c * rho^-ii
        dim3 grd(BN_TOK / 128, BN_TOK / 64, H_NUM);
        wmma_gemm_f32<false, 2><<<grd, blk, 0, stream>>>(
            Bmat, BN_TOK, (long)BN_TOK * BN_TOK,
            cur,  BN_TOK, (long)BN_TOK * BN_TOK,
            nxt,  BN_TOK, (long)BN_TOK * BN_TOK,
            BN_TOK, 2, 1.0f, nullptr, nullptr, nullptr);
        axpy_pow_kernel<<<dim3((unsigned)(NEL / 256)), blk, 0, stream>>>(
            nxt, rhos, Lg, c, ii, BN_TOK);
        float* tmp = cur; cur = nxt; nxt = tmp;
    }

    // 6) L1 row norms of (diag - rho^g * Lg), then combined attention matrix
    lrow_kernel<<<dim3(H_NUM * BN_TOK), blk, 0, stream>>>(Lg, rhos, rsL, BN_TOK);
    combine_kernel<<<dim3((unsigned)(NEL / 256)), blk, 0, stream>>>(
        WG, Lg, deg, rhos, rsL, Bmat, BN_TOK);

    // 7) attn = comb @ V_h -> [H, BN, 64]; flat layout == output flat layout
    //    (N=64 -> WS=1 variant, 32x64 block tiles)
    {
        dim3 grd(HD_DIM / 64, BN_TOK / 32, H_NUM);
        wmma_gemm_f32<false, 1><<<grd, blk, 0, stream>>>(
            Bmat, BN_TOK, (long)BN_TOK * BN_TOK,
            v,    E_DIM,  HD_DIM,
            out,  HD_DIM, (long)BN_TOK * HD_DIM,
            BN_TOK, 2, 1.0f, nullptr, nullptr, nullptr);
    }
}